// LossFn_197568495817
// MI455X (gfx1250) — compile-verified
//
#include <hip/hip_runtime.h>
#include <hip/hip_bf16.h>
#include <math.h>

// ---------------- problem constants ----------------
#define Bn 4
#define Tn 2048
#define Cn 48
#define Dn 512
#define Ln 16
#define Kn 9
#define CSn 15
#define NTR 15          // L-1
#define NCAND 45        // 3*(L-1)
#define Wn 31           // 2*NTR+1
#define WINL 19         // int(T/L*0.15)
#define ALPHA_ 1.0f
#define BETA_  1.0f
#define GAMMA_ 0.1f
#define TEMP_  0.1f
#define LN2F 0.6931471805599453f
#define INFF __builtin_inff()

typedef float v2f __attribute__((ext_vector_type(2)));
typedef float v8f __attribute__((ext_vector_type(8)));

// ---------------- helpers ----------------
__device__ __forceinline__ float log_sigmoid(float x) {
    return fminf(x, 0.0f) - log1pf(expf(-fabsf(x)));
}

// ---------------- K1: tok_loss ----------------
__global__ void k_tok(const float* __restrict__ tok_logit,
                      const float* __restrict__ vid, float* __restrict__ scal) {
    __shared__ float red[256];
    int i = threadIdx.x;
    float v = 0.0f;
    if (i < Bn * Cn) {
        float x = tok_logit[i], y = vid[i];
        v = -(y * log_sigmoid(x) + (1.0f - y) * log_sigmoid(-x));
    }
    red[i] = v; __syncthreads();
    for (int s = 128; s; s >>= 1) { if (i < s) red[i] += red[i + s]; __syncthreads(); }
    if (i == 0) scal[0] = red[0] / (float)(Bn * Cn);
}

// ---------------- K2: softmax rows + logZ + row entropy H = sum p*log p ----------------
__global__ void k_softmax(const float* __restrict__ fr, float* __restrict__ prob,
                          float* __restrict__ logZ, float* __restrict__ ent) {
    int row = blockIdx.x * blockDim.x + threadIdx.x;
    if (row >= Bn * Tn) return;
    const float* x = fr + (size_t)row * Cn;
    float mx = -INFF;
    for (int c = 0; c < Cn; ++c) mx = fmaxf(mx, x[c]);
    float s = 0.0f;
    for (int c = 0; c < Cn; ++c) s += expf(x[c] - mx);
    float lz = mx + logf(s);
    float* p = prob + (size_t)row * Cn;
    float h = 0.0f;
    for (int c = 0; c < Cn; ++c) {
        float lp = x[c] - lz;          // log p, free
        float pv = expf(lp);
        p[c] = pv;
        h += pv * lp;                  // H = sum p log p (no extra logf)
    }
    logZ[row] = lz;
    ent[row] = h;
}

// ---------------- K3: boundary (JS) score via entropy identity ----------------
// KL(a||m)+KL(b||m) = H(a)+H(b) - sum (va+vb)*log(0.5*(va+vb)+1e-32)
__global__ void k_bdy(const float* __restrict__ prob, const float* __restrict__ ent,
                      float* __restrict__ score) {
    int row = blockIdx.x * blockDim.x + threadIdx.x;
    if (row >= Bn * Tn) return;
    int b = row / Tn, t = row % Tn;
    const float* Pb = prob + (size_t)b * Tn * Cn;
    const float* Hb_ = ent + (size_t)b * Tn;
    float acc = 8.0f;                              // diagonal: 8 entries of +1, js==1
    const int c = Kn / 2;
    for (int i = 0; i < Kn; ++i) {
        int ra = t + i - c;
        bool ina = (ra >= 0) && (ra < Tn);
        const float* pa = Pb + (size_t)(ina ? ra : 0) * Cn;
        float Ha = ina ? Hb_[ra] : 0.0f;
        for (int j = i + 1; j < Kn; ++j) {
            int rb = t + j - c;
            bool inb = (rb >= 0) && (rb < Tn);
            const float* pb = Pb + (size_t)(inb ? rb : 0) * Cn;
            float Hbv = inb ? Hb_[rb] : 0.0f;
            float w = ((i < c) == (j < c)) ? 1.0f : -1.0f;
            float sm = 0.0f;
            for (int d = 0; d < Cn; ++d) {
                float va = ina ? pa[d] : 0.0f;
                float vb = inb ? pb[d] : 0.0f;
                float sv = va + vb;
                sm += sv * logf(0.5f * sv + 1e-32f);   // sv==0 -> 0 * finite = 0
            }
            float kl2 = Ha + Hbv - sm;                 // klab + klba
            float js = 1.0f - kl2 / LN2F;
            acc += 2.0f * w * js;
        }
    }
    score[row] = (t == 0) ? -INFF : acc / 81.0f;
}

// ---------------- K4: pick 45 candidates (argmax + suppression), sort ----------------
__global__ void k_pick(const float* __restrict__ score, int* __restrict__ cand) {
    int b = blockIdx.x, tid = threadIdx.x;
    __shared__ float s[Tn];
    __shared__ float rv[256];
    __shared__ int   ri[256];
    __shared__ int   cl[NCAND];
    for (int t = tid; t < Tn; t += 256) s[t] = score[b * Tn + t];
    __syncthreads();
    for (int it = 0; it < NCAND; ++it) {
        float bv = -INFF; int bi = Tn;
        for (int t = tid; t < Tn; t += 256) {
            float v = s[t];
            if (v > bv) { bv = v; bi = t; }     // first (lowest t) on ties
        }
        rv[tid] = bv; ri[tid] = bi; __syncthreads();
        for (int off = 128; off; off >>= 1) {
            if (tid < off) {
                if (rv[tid + off] > rv[tid] ||
                    (rv[tid + off] == rv[tid] && ri[tid + off] < ri[tid])) {
                    rv[tid] = rv[tid + off]; ri[tid] = ri[tid + off];
                }
            }
            __syncthreads();
        }
        int imax = ri[0];
        if (tid == 0) cl[it] = imax;
        __syncthreads();
        for (int t = tid; t < Tn; t += 256)
            if (abs(t - imax) <= WINL) s[t] = -INFF;
        __syncthreads();
    }
    if (tid == 0) {
        for (int i = 1; i < NCAND; ++i) {
            int key = cl[i], j = i - 1;
            while (j >= 0 && cl[j] > key) { cl[j + 1] = cl[j]; --j; }
            cl[j + 1] = key;
        }
        for (int i = 0; i < NCAND; ++i) cand[b * NCAND + i] = cl[i];
    }
}

// ---------------- K5a: cls scores (15 x 45 per batch) ----------------
__global__ void k_cls(const float* __restrict__ prob, const float* __restrict__ score,
                      const int* __restrict__ cand, const int* __restrict__ transcript,
                      float* __restrict__ cls) {
    int b = blockIdx.x;
    const float* Pb = prob + (size_t)b * Tn * Cn;
    for (int idx = threadIdx.x; idx < NTR * NCAND; idx += blockDim.x) {
        int j = idx / NCAND, i = idx % NCAND;
        int ci = cand[b * NCAND + i];
        int c1 = transcript[b * Ln + j];
        int c2 = transcript[b * Ln + j + 1];
        float s = 0.0f;
        for (int k = 0; k < CSn; ++k) {
            int t = ci + k - CSn / 2;
            float p1 = 0.0f, p2 = 0.0f;
            if (t >= 0 && t < Tn) {
                p1 = Pb[(size_t)t * Cn + c1];
                p2 = Pb[(size_t)t * Cn + c2];
            }
            float k0 = (k < CSn / 2) ? 1.0f : -1.0f;
            s += k0 * (p1 - p2);
        }
        cls[(b * NTR + j) * NCAND + i] = s / (2.0f * CSn) + score[b * Tn + ci];
    }
}

// ---------------- K5b: DP + backtrack -> bdy; sorted labels ----------------
__global__ void k_dp(const float* __restrict__ cls, const int* __restrict__ cand,
                     const int* __restrict__ transcript, int* __restrict__ bdy,
                     int* __restrict__ labels) {
    int b = blockIdx.x;
    __shared__ signed char dir[NCAND][Wn];
    if (threadIdx.x != 0) return;
    const float* cb = cls + (size_t)b * NTR * NCAND;
    float prev[Wn], nxt[Wn];
    for (int j = 0; j < Wn; ++j) { prev[j] = INFF; dir[0][j] = 0; }
    prev[0] = 0.0f;
    prev[1] = -cb[0 * NCAND + 0];
    dir[0][1] = 1;
    for (int ii = 1; ii < NCAND; ++ii) {
        for (int j = 0; j < Wn; ++j) {
            float dm1 = (j >= 1) ? prev[j - 1] : INFF;
            float dm2 = (j >= 2) ? prev[j - 2] : INFF;
            int ta = j / 2; if (ta > NTR - 1) ta = NTR - 1;
            float ca = -cb[ta * NCAND + ii];
            float nv; int nd;
            if (j >= 2) {
                if ((j & 1) == 0) { nv = fminf(prev[j], dm1); nd = (prev[j] < dm1) ? 0 : 1; }
                else              { nv = ca + fminf(dm1, dm2); nd = (dm1 < dm2) ? 1 : 2; }
            } else { nv = INFF; nd = 0; }
            if (j == 0) { nv = (ii < NCAND - NTR) ? 0.0f : INFF; nd = 0; }
            if (j == 1) {
                nv = (ii <= NCAND - NTR) ? -cb[0 * NCAND + ii] : INFF;
                nd = (ii <= NCAND - NTR) ? 1 : 0;
            }
            nxt[j] = nv; dir[ii][j] = (signed char)nd;
        }
        for (int j = 0; j < Wn; ++j) prev[j] = nxt[j];
    }
    int cur = (prev[2 * NTR] < prev[2 * NTR - 1]) ? (2 * NTR) : (2 * NTR - 1);
    int r[NTR];
    for (int j = 0; j < NTR; ++j) r[j] = 0;
    for (int ii = NCAND - 1; ii >= 0; --ii) {
        if (cur & 1) { int oi = cur / 2; if (oi < NTR) r[oi] = ii; }
        cur -= dir[ii][cur];
    }
    for (int j = 0; j < NTR; ++j)
        bdy[b * NTR + j] = cand[b * NCAND + r[j]];
    int lab[Ln];
    for (int l = 0; l < Ln; ++l) lab[l] = transcript[b * Ln + l];
    for (int i = 1; i < Ln; ++i) {
        int key = lab[i], j = i - 1;
        while (j >= 0 && lab[j] > key) { lab[j + 1] = lab[j]; --j; }
        lab[j + 1] = key;
    }
    for (int l = 0; l < Ln; ++l) labels[b * Ln + l] = lab[l];
}

// ---------------- K5c: pseudo labels + sorted-label slot ----------------
__global__ void k_pse(const int* __restrict__ bdy, const int* __restrict__ transcript,
                      const int* __restrict__ labels, int* __restrict__ pse,
                      int* __restrict__ slot) {
    int row = blockIdx.x * blockDim.x + threadIdx.x;
    if (row >= Bn * Tn) return;
    int b = row / Tn, t = row % Tn;
    int seg = 0;
    for (int j = 0; j < NTR; ++j) seg += (t >= bdy[b * NTR + j]) ? 1 : 0;
    int lbl = transcript[b * Ln + seg];
    pse[row] = lbl;
    int sl = 0;
    for (int l = 0; l < Ln; ++l) if (labels[b * Ln + l] == lbl) sl = l;
    slot[row] = sl;
}

// ---------------- K6: fr_loss sums ----------------
__global__ void k_fr(const float* __restrict__ fr_logit, const float* __restrict__ logZ,
                     const int* __restrict__ pse, const unsigned char* __restrict__ mask,
                     float* __restrict__ scal) {
    __shared__ float r1[256], r2[256];
    int tid = threadIdx.x;
    float sc = 0.0f, sm = 0.0f;
    for (int row = tid; row < Bn * Tn; row += 256) {
        float m = mask[row] ? 1.0f : 0.0f;
        float ce = logZ[row] - fr_logit[(size_t)row * Cn + pse[row]];
        sc += ce * m; sm += m;
    }
    r1[tid] = sc; r2[tid] = sm; __syncthreads();
    for (int s = 128; s; s >>= 1) {
        if (tid < s) { r1[tid] += r1[tid + s]; r2[tid] += r2[tid + s]; }
        __syncthreads();
    }
    if (tid == 0) { scal[1] = r1[0]; scal[2] = r2[0]; }
}

// ---------------- K7a: normalized token features ----------------
__global__ void k_tk(const float* __restrict__ feat, float* __restrict__ tkN) {
    int b = blockIdx.x / Cn, c = blockIdx.x % Cn, tid = threadIdx.x;
    __shared__ float red[256];
    const float* rowp = feat + ((size_t)b * (Cn + Tn) + c) * Dn;
    float v0 = rowp[tid], v1 = rowp[tid + 256];
    red[tid] = v0 * v0 + v1 * v1; __syncthreads();
    for (int s = 128; s; s >>= 1) { if (tid < s) red[tid] += red[tid + s]; __syncthreads(); }
    float nrm = fmaxf(sqrtf(red[0]), 1e-12f);
    float* o = tkN + ((size_t)b * Cn + c) * Dn;
    o[tid] = v0 / nrm; o[tid + 256] = v1 / nrm;
}

// ---------------- K7b: segment-mean la + normalize ----------------
__global__ void k_la(const float* __restrict__ feat, const int* __restrict__ slot,
                     const unsigned char* __restrict__ mask, float* __restrict__ laN) {
    int b = blockIdx.x / Ln, l = blockIdx.x % Ln, tid = threadIdx.x;
    __shared__ unsigned char selS[Tn];
    __shared__ float red[256];
    float cp = 0.0f;
    for (int t = tid; t < Tn; t += 256) {
        unsigned char sel = (slot[b * Tn + t] == l && mask[b * Tn + t]) ? 1 : 0;
        selS[t] = sel; cp += (float)sel;
    }
    red[tid] = cp; __syncthreads();
    for (int s = 128; s; s >>= 1) { if (tid < s) red[tid] += red[tid + s]; __syncthreads(); }
    float cnt = fmaxf(red[0], 1.0f);
    __syncthreads();
    const float* frf = feat + ((size_t)b * (Cn + Tn) + Cn) * Dn;
    float a0 = 0.0f, a1 = 0.0f;
    for (int t = 0; t < Tn; ++t) {
        if (selS[t]) { a0 += frf[(size_t)t * Dn + tid]; a1 += frf[(size_t)t * Dn + tid + 256]; }
    }
    a0 /= cnt; a1 /= cnt;
    red[tid] = a0 * a0 + a1 * a1; __syncthreads();
    for (int s = 128; s; s >>= 1) { if (tid < s) red[tid] += red[tid + s]; __syncthreads(); }
    float nrm = fmaxf(sqrtf(red[0]), 1e-12f);
    float* o = laN + ((size_t)b * Ln + l) * Dn;
    o[tid] = a0 / nrm; o[tid + 256] = a1 / nrm;
}

// ---------------- K8: sim = la_norm @ tk_norm^T, async-LDS staged WMMA ----------------
// One wave per 16x16 output tile. Both 16x512 f32 tiles (32 KB each) are staged into
// LDS with GLOBAL_LOAD_ASYNC_TO_LDS_B128 (32 lanes x 16 B = 512 B per instruction,
// ASYNCcnt-tracked), then the 128-step K loop feeds V_WMMA_F32_16X16X4_F32 from LDS
// (ds_load_b64) with no per-iteration global-latency stalls.
__global__ void k_sim(const float* __restrict__ laN, const float* __restrict__ tkN,
                      float* __restrict__ sim) {
    __shared__ __align__(16) float lA[Ln * Dn];   // 32 KB
    __shared__ __align__(16) float lB[16 * Dn];   // 32 KB
    int b = blockIdx.x / 3, tile = blockIdx.x % 3;
    int lane = threadIdx.x & 31;
    const char* gA = (const char*)(laN + (size_t)b * Ln * Dn);
    const char* gB = (const char*)(tkN + ((size_t)b * Cn + tile * 16) * Dn);
    unsigned ldsA = (unsigned)(uintptr_t)(void*)lA;   // low 32 bits = LDS byte address
    unsigned ldsB = (unsigned)(uintptr_t)(void*)lB;
    // 32 KB / (32 lanes * 16 B) = 64 iterations per matrix
    for (int i = 0; i < 64; ++i) {
        unsigned off16 = (unsigned)((i * 32 + lane) * 16);
        unsigned long long ga = (unsigned long long)(uintptr_t)gA + off16;
        unsigned long long gb = (unsigned long long)(uintptr_t)gB + off16;
        unsigned da = ldsA + off16;
        unsigned db = ldsB + off16;
        asm volatile("global_load_async_to_lds_b128 %0, %1, off"
                     :: "v"(da), "v"(ga) : "memory");
        asm volatile("global_load_async_to_lds_b128 %0, %1, off"
                     :: "v"(db), "v"(gb) : "memory");
    }
    asm volatile("s_wait_asynccnt 0x0" ::: "memory");
    __syncthreads();

    int half = lane >> 4;     // 0: K pair (0,1)  1: K pair (2,3)
    int r    = lane & 15;     // M (A row) and N (B column) index
    const float* arow = lA + r * Dn;
    const float* brow = lB + r * Dn;
    v8f c = {0.f, 0.f, 0.f, 0.f, 0.f, 0.f, 0.f, 0.f};
    for (int k = 0; k < Dn; k += 4) {
        v2f a  = *(const v2f*)(arow + k + 2 * half);
        v2f bb = *(const v2f*)(brow + k + 2 * half);
        c = __builtin_amdgcn_wmma_f32_16x16x4_f32(false, a, false, bb,
                                                  (short)0, c, false, false);
    }
    float* sb = sim + (size_t)b * Ln * Cn;
    for (int j = 0; j < 8; ++j) {
        int m = j + 8 * half;                       // C/D layout: vgpr j, lane-half -> row
        sb[m * Cn + tile * 16 + r] = c[j] * (1.0f / TEMP_);
    }
}

// ---------------- K9: glc log-softmax + combine ----------------
__global__ void k_final(const float* __restrict__ sim, const int* __restrict__ labels,
                        const float* __restrict__ scal, float* __restrict__ out) {
    __shared__ float red[64];
    int tid = threadIdx.x;                          // tid = b*16 + m
    const float* srow = sim + (size_t)tid * Cn;
    float mx = -INFF;
    for (int cidx = 0; cidx < Cn; ++cidx) mx = fmaxf(mx, srow[cidx]);
    float s = 0.0f;
    for (int cidx = 0; cidx < Cn; ++cidx) s += expf(srow[cidx] - mx);
    float lz = mx + logf(s);
    int lbl = labels[tid];
    red[tid] = -(srow[lbl] - lz);
    __syncthreads();
    for (int off = 32; off; off >>= 1) { if (tid < off) red[tid] += red[tid + off]; __syncthreads(); }
    if (tid == 0) {
        float glc = red[0] / 64.0f;
        out[0] = ALPHA_ * scal[0] + BETA_ * (scal[1] / fmaxf(scal[2], 1e-12f)) + GAMMA_ * glc;
    }
}

// ---------------- launch ----------------
extern "C" void kernel_launch(void* const* d_in, const int* in_sizes, int n_in,
                              void* d_out, int out_size, void* d_ws, size_t ws_size,
                              hipStream_t stream) {
    (void)in_sizes; (void)n_in; (void)out_size; (void)ws_size;
    // inputs: epoch, tok_logit, fr_logit, mask, transcript, vid_multi_hot, feat
    const float*         tok_logit  = (const float*)d_in[1];
    const float*         fr_logit   = (const float*)d_in[2];
    const unsigned char* mask       = (const unsigned char*)d_in[3];
    const int*           transcript = (const int*)d_in[4];
    const float*         vid        = (const float*)d_in[5];
    const float*         feat       = (const float*)d_in[6];
    float* out = (float*)d_out;

    char* w = (char*)d_ws;
    size_t off = 0;
    auto carve = [&](size_t bytes) -> void* {
        void* p = w + off;
        off += (bytes + 255) & ~(size_t)255;
        return p;
    };
    float* prob   = (float*)carve((size_t)Bn * Tn * Cn * 4);
    float* logZ   = (float*)carve((size_t)Bn * Tn * 4);
    float* ent    = (float*)carve((size_t)Bn * Tn * 4);
    float* score  = (float*)carve((size_t)Bn * Tn * 4);
    float* cls    = (float*)carve((size_t)Bn * NTR * NCAND * 4);
    int*   cand   = (int*)  carve((size_t)Bn * NCAND * 4);
    int*   bdy    = (int*)  carve((size_t)Bn * NTR * 4);
    int*   labels = (int*)  carve((size_t)Bn * Ln * 4);
    int*   pse    = (int*)  carve((size_t)Bn * Tn * 4);
    int*   slot   = (int*)  carve((size_t)Bn * Tn * 4);
    float* laN    = (float*)carve((size_t)Bn * Ln * Dn * 4);
    float* tkN    = (float*)carve((size_t)Bn * Cn * Dn * 4);
    float* sim    = (float*)carve((size_t)Bn * Ln * Cn * 4);
    float* scal   = (float*)carve(8 * 4);

    const int rows = Bn * Tn;
    k_tok    <<<1, 256, 0, stream>>>(tok_logit, vid, scal);
    k_softmax<<<(rows + 255) / 256, 256, 0, stream>>>(fr_logit, prob, logZ, ent);
    k_bdy    <<<(rows + 255) / 256, 256, 0, stream>>>(prob, ent, score);
    k_pick   <<<Bn, 256, 0, stream>>>(score, cand);
    k_cls    <<<Bn, 256, 0, stream>>>(prob, score, cand, transcript, cls);
    k_dp     <<<Bn, 32, 0, stream>>>(cls, cand, transcript, bdy, labels);
    k_pse    <<<(rows + 255) / 256, 256, 0, stream>>>(bdy, transcript, labels, pse, slot);
    k_fr     <<<1, 256, 0, stream>>>(fr_logit, logZ, pse, mask, scal);
    k_tk     <<<Bn * Cn, 256, 0, stream>>>(feat, tkN);
    k_la     <<<Bn * Ln, 256, 0, stream>>>(feat, slot, mask, laN);
    k_sim    <<<Bn * 3, 32, 0, stream>>>(laN, tkN, sim);
    k_final  <<<1, 64, 0, stream>>>(sim, labels, scal, out);
}